// _GraphConvolutionLayer_23742579212562
// MI455X (gfx1250) — compile-verified
//
#include <hip/hip_runtime.h>
#include <hip/hip_bf16.h>

typedef __attribute__((ext_vector_type(2))) float v2f;
typedef __attribute__((ext_vector_type(8))) float v8f;

#define DDIM   128   // feature dim
#define KNN    32    // neighbors per node
#define TILE_N 16    // output rows per block
#define WPAD   130   // padded LDS row stride in floats (even -> 8B aligned, +2 kills bank conflicts)
#define THREADS 256  // 8 wave32

__global__ __launch_bounds__(THREADS)
void gcl_fused_kernel(const float* __restrict__ Input,
                      const int*   __restrict__ Idx,
                      const float* __restrict__ C,
                      const float* __restrict__ W,
                      const float* __restrict__ b,
                      float*       __restrict__ out,
                      int N)
{
    extern __shared__ float lds[];
    float* wLDS = lds;                        // DDIM * WPAD floats  (W, row-major [e][d], padded)
    float* yLDS = lds + DDIM * WPAD;          // TILE_N * WPAD floats (weighted-gather tile)
    float* sLDS = yLDS + TILE_N * WPAD;       // TILE_N floats       (row sums of C, for bias)

    const int t  = threadIdx.x;
    const int n0 = blockIdx.x * TILE_N;

    // ---- Load W into LDS (coalesced, padded rows) ----
    for (int i = t; i < DDIM * DDIM; i += THREADS) {
        const int r = i >> 7;        // i / 128
        const int c = i & (DDIM - 1);
        wLDS[r * WPAD + c] = W[i];
    }

    // ---- Stage 1: weighted gather-reduce  y[n,d] = sum_k C[n,k] * Input[I[n,k], d] ----
    // thread t owns column d = t & 127 for 8 rows: [half*8, half*8+8)
    const int d    = t & (DDIM - 1);
    const int half = t >> 7;         // 0 or 1

    float acc[8];
#pragma unroll
    for (int r = 0; r < 8; ++r) acc[r] = 0.0f;

    for (int k = 0; k < KNN; ++k) {
#pragma unroll
        for (int r = 0; r < 8; ++r) {
            const int n = n0 + half * 8 + r;
            if (n < N) {
                const int   gi = Idx[n * KNN + k];   // same value for all 128 d-threads -> broadcast
                const float cw = C[n * KNN + k];
                acc[r] += cw * Input[(long)gi * DDIM + d];
            }
        }
    }
#pragma unroll
    for (int r = 0; r < 8; ++r)
        yLDS[(half * 8 + r) * WPAD + d] = acc[r];

    // per-row sum of C (scales the bias since o = xW^T + b is summed with weights C)
    if (t < TILE_N) {
        const int n = n0 + t;
        float s = 0.0f;
        if (n < N)
            for (int k = 0; k < KNN; ++k) s += C[n * KNN + k];
        sLDS[t] = s;
    }
    __syncthreads();

    // ---- Stage 2: WMMA fp32 GEMM  out_tile[16,16] = y_tile[16,128] x W[e0:e0+16, :]^T ----
    // wave w handles output columns e in [16w, 16w+16)
    const int wave = t >> 5;
    const int lane = t & 31;
    const int hi   = lane >> 4;      // 0: K pair {0,1}, 1: K pair {2,3} within each 16x4 step
    const int lo   = lane & 15;      // M index for A, N index for B
    const int e0   = wave * 16;

    v8f cacc = {};
#pragma unroll
    for (int k0 = 0; k0 < DDIM; k0 += 4) {
        // A fragment: 16x4 fp32, lane(lo,hi) holds y[lo][k0+2*hi], y[lo][k0+2*hi+1]
        const float* ap = &yLDS[lo * WPAD + k0 + 2 * hi];
        // B fragment: 4x16 fp32 of W^T, i.e. B[k][e] = W[e][k]
        const float* bp = &wLDS[(e0 + lo) * WPAD + k0 + 2 * hi];
        v2f av; av.x = ap[0]; av.y = ap[1];
        v2f bv; bv.x = bp[0]; bv.y = bp[1];
        cacc = __builtin_amdgcn_wmma_f32_16x16x4_f32(
            /*neg_a=*/false, av, /*neg_b=*/false, bv,
            /*c_mod=*/(short)0, cacc, /*reuse_a=*/false, /*reuse_b=*/false);
    }

    // ---- Epilogue: + s[n]*b[e], sigmoid, store ----
    const float bias = b[e0 + lo];
#pragma unroll
    for (int v = 0; v < 8; ++v) {
        const int nl = v + 8 * hi;       // C/D layout: VGPR v -> M = v + 8*(lane>=16)
        const int n  = n0 + nl;
        if (n < N) {
            const float val = cacc[v] + sLDS[nl] * bias;
            out[(long)n * DDIM + (e0 + lo)] = 1.0f / (1.0f + __expf(-val));
        }
    }
}

extern "C" void kernel_launch(void* const* d_in, const int* in_sizes, int n_in,
                              void* d_out, int out_size, void* d_ws, size_t ws_size,
                              hipStream_t stream)
{
    const float* Input = (const float*)d_in[0];   // [M, 128] f32
    const int*   Idx   = (const int*)  d_in[1];   // [N, 32]  int
    const float* C     = (const float*)d_in[2];   // [N, 32]  f32
    const float* W     = (const float*)d_in[3];   // [128,128] f32 (row-major [e][d])
    const float* b     = (const float*)d_in[4];   // [128] f32
    float*       out   = (float*)d_out;           // [N, 128] f32

    const int N = in_sizes[1] / KNN;
    const int blocks = (N + TILE_N - 1) / TILE_N;
    const size_t shmem = (size_t)(DDIM * WPAD + TILE_N * WPAD + TILE_N) * sizeof(float);

    hipLaunchKernelGGL(gcl_fused_kernel, dim3(blocks), dim3(THREADS), shmem, stream,
                       Input, Idx, C, W, b, out, N);
}